// pos_Bi_Attention_90950227460840
// MI455X (gfx1250) — compile-verified
//
#include <hip/hip_runtime.h>
#include <hip/hip_bf16.h>
#include <math.h>

// ---------------- problem constants (match reference) ----------------
#define HH    320
#define WWID  320
#define HWSZ  (HH * WWID)
#define CC    32
#define NPTS  20000
#define BB    4
#define NSH   9   // 9 neighbor shifts
#define NHD   2   // heads
#define HD    16  // head dim

typedef __attribute__((ext_vector_type(2))) float v2f;
typedef __attribute__((ext_vector_type(8))) float v8f;

__device__ __constant__ int c_shift[NSH][2] = {
    {0, 0}, {-1, 0}, {1, 0}, {0, 1}, {-1, 1}, {1, 1}, {0, -1}, {-1, -1}, {1, -1}};

// D = A(16x4,f32) * B(4x16,f32) + C  -> 16x16 f32, full fp32 precision
__device__ __forceinline__ v8f wmma4(v2f a, v2f b, v8f c) {
    return __builtin_amdgcn_wmma_f32_16x16x4_f32(
        /*neg_a=*/false, a, /*neg_b=*/false, b,
        /*c_mod=*/(short)0, c, /*reuse_a=*/false, /*reuse_b=*/false);
}

// ---------------- grid build: grid[(dir*B+b)*HW + r*W+c] = n ----------------
__global__ void build_grid_kernel(const int* __restrict__ li_c,
                                  const int* __restrict__ ra_c,
                                  int* __restrict__ grid) {
    int t = blockIdx.x * blockDim.x + threadIdx.x;
    if (t >= 2 * BB * NPTS) return;
    int dir = t / (BB * NPTS);
    int rem = t - dir * (BB * NPTS);
    int b = rem / NPTS;
    int n = rem - b * NPTS;
    // dir 0: queries=li, kv=ra  |  dir 1: queries=ra, kv=li
    const int* c = dir ? li_c : ra_c;
    int c0 = c[(b * NPTS + n) * 2 + 0];
    int c1 = c[(b * NPTS + n) * 2 + 1];
    grid[(size_t)(dir * BB + b) * HWSZ + c0 * WWID + c1] = n;
}

// ---------------- fused windowed cross-attention (WMMA) ----------------
// grid: x = tile-of-64-points, y = batch, z = direction; 128 threads = 4 waves,
// each wave32 owns one 16-point WMMA tile.
__global__ __launch_bounds__(128)
void fused_attn_kernel(const float* __restrict__ li_f, const int* __restrict__ li_c,
                       const float* __restrict__ ra_f, const int* __restrict__ ra_c,
                       const float* __restrict__ pos,
                       const float* __restrict__ inw1, const float* __restrict__ inb1,
                       const float* __restrict__ outw1, const float* __restrict__ outb1,
                       const float* __restrict__ inw2, const float* __restrict__ inb2,
                       const float* __restrict__ outw2, const float* __restrict__ outb2,
                       const int* __restrict__ grid_ws,
                       float* __restrict__ out) {
    __shared__ float s_win[96 * 32];     // [wq;wk;wv] row-major (3C x C)
    __shared__ float s_wout[32 * 32];    // out_w (C x C)
    __shared__ float s_bin[96];
    __shared__ float s_bout[32];
    __shared__ float s_pos[NSH * 32];
    __shared__ float s_q[4][16 * 32];    // qh per wave
    __shared__ float s_kh[4][16 * 32];   // kh scratch per wave (per shift)
    __shared__ float s_o[4][16 * 32];    // attention output per wave
    __shared__ float s_attn[4][NHD * 16 * NSH];
    __shared__ int   s_sel[4][NSH * 16];
    __shared__ int   s_pix[4][16];

    const int dir = blockIdx.z;
    const int b   = blockIdx.y;
    const int tid = threadIdx.x;
    const int w   = tid >> 5;
    const int lane = tid & 31;

    const float* qf  = dir ? ra_f : li_f;
    const int*   qc  = dir ? ra_c : li_c;
    const float* kf  = dir ? li_f : ra_f;
    const float* inw = dir ? inw2 : inw1;
    const float* inb = dir ? inb2 : inb1;
    const float* onw = dir ? outw2 : outw1;
    const float* onb = dir ? outb2 : outb1;
    const int* grid = grid_ws + (size_t)(dir * BB + b) * HWSZ;

    // ---- stage block-shared constants into LDS ----
    for (int i = tid; i < 96 * 32; i += 128) s_win[i] = inw[i];
    for (int i = tid; i < 32 * 32; i += 128) s_wout[i] = onw[i];
    for (int i = tid; i < NSH * 32; i += 128) s_pos[i] = pos[i];
    if (tid < 96) s_bin[tid] = inb[tid];
    if (tid < 32) s_bout[tid] = onb[tid];

    const int n0 = (blockIdx.x * 4 + w) * 16;

    // ---- per-tile pixel targets + neighbor kv indices ----
    if (lane < 16) {
        int n = n0 + lane;
        int p = -1;
        if (n < NPTS) {
            int c0 = qc[(b * NPTS + n) * 2 + 0];
            int c1 = qc[(b * NPTS + n) * 2 + 1];
            p = c0 * WWID + c1;
        }
        s_pix[w][lane] = p;
    }
    for (int e = lane; e < NSH * 16; e += 32) {
        int l = e >> 4, m = e & 15;
        int n = n0 + m;
        int sel = -1;
        if (n < NPTS) {
            int c0 = qc[(b * NPTS + n) * 2 + 0] + c_shift[l][0];
            int c1 = qc[(b * NPTS + n) * 2 + 1] + c_shift[l][1];
            if (c0 >= 0 && c0 < HH && c1 >= 0 && c1 < WWID) sel = grid[c0 * WWID + c1];
        }
        s_sel[w][e] = sel;
    }
    __syncthreads();

    // WMMA lane roles: mrow = A-row / B-col / D-col selector, hi = K-subpair
    const int mrow = lane & 15;
    const int hi   = lane >> 4;

    int nq = n0 + mrow;
    if (nq >= NPTS) nq = NPTS - 1;  // safe clamp; stores gated by s_pix
    const float* qrow = qf + ((size_t)b * NPTS + nq) * CC;

    // ---- Q projection: qh = q @ wq^T + bq  (two 16-col tiles == heads) ----
    v8f acc0, acc1;
    {
        float bq0 = s_bin[mrow], bq1 = s_bin[16 + mrow];
        #pragma unroll
        for (int r = 0; r < 8; ++r) { acc0[r] = bq0; acc1[r] = bq1; }
        const float* wb0 = &s_win[(0 + mrow) * 32];
        const float* wb1 = &s_win[(16 + mrow) * 32];
        #pragma unroll
        for (int kk = 0; kk < 8; ++kk) {
            int kb = kk * 4 + hi * 2;
            v2f a;  a[0] = qrow[kb];  a[1] = qrow[kb + 1];
            v2f b0; b0[0] = wb0[kb];  b0[1] = wb0[kb + 1];
            v2f b1; b1[0] = wb1[kb];  b1[1] = wb1[kb + 1];
            acc0 = wmma4(a, b0, acc0);
            acc1 = wmma4(a, b1, acc1);
        }
    }
    #pragma unroll
    for (int r = 0; r < 8; ++r) {
        int m = r + 8 * hi;
        s_q[w][m * 32 + mrow]      = acc0[r];
        s_q[w][m * 32 + 16 + mrow] = acc1[r];
    }
    __syncthreads();

    // score lanes: lane = hi*16 + mrow  ->  (head=hi, point=mrow)
    float qreg[HD];
    #pragma unroll
    for (int d = 0; d < HD; ++d) qreg[d] = s_q[w][mrow * 32 + hi * HD + d];

    // ---- pass 1: K projection per shift -> scores ----
    float sc[NSH];
    {
        const float* wb0 = &s_win[(32 + mrow) * 32];
        const float* wb1 = &s_win[(48 + mrow) * 32];
        float bk0 = s_bin[32 + mrow], bk1 = s_bin[48 + mrow];
        for (int l = 0; l < NSH; ++l) {
            int sel = s_sel[w][l * 16 + mrow];
            const float* krow = kf + ((size_t)b * NPTS + (sel < 0 ? 0 : sel)) * CC;
            v8f k0, k1;
            #pragma unroll
            for (int r = 0; r < 8; ++r) { k0[r] = bk0; k1[r] = bk1; }
            #pragma unroll
            for (int kk = 0; kk < 8; ++kk) {
                int kb = kk * 4 + hi * 2;
                v2f a;
                a[0] = (sel >= 0) ? krow[kb]     : 0.0f;
                a[1] = (sel >= 0) ? krow[kb + 1] : 0.0f;
                v2f b0; b0[0] = wb0[kb]; b0[1] = wb0[kb + 1];
                v2f b1; b1[0] = wb1[kb]; b1[1] = wb1[kb + 1];
                k0 = wmma4(a, b0, k0);
                k1 = wmma4(a, b1, k1);
            }
            #pragma unroll
            for (int r = 0; r < 8; ++r) {
                int m = r + 8 * hi;
                s_kh[w][m * 32 + mrow]      = k0[r];
                s_kh[w][m * 32 + 16 + mrow] = k1[r];
            }
            __syncthreads();
            float s = 0.0f;
            #pragma unroll
            for (int d = 0; d < HD; ++d) s += qreg[d] * s_kh[w][mrow * 32 + hi * HD + d];
            sc[l] = s * 0.25f;  // 1/sqrt(HD)
            __syncthreads();
        }
    }

    // ---- softmax over the 9 shifts (invalid entries included, as in ref) ----
    {
        float mx = sc[0];
        #pragma unroll
        for (int l = 1; l < NSH; ++l) mx = fmaxf(mx, sc[l]);
        float sum = 0.0f;
        #pragma unroll
        for (int l = 0; l < NSH; ++l) { sc[l] = expf(sc[l] - mx); sum += sc[l]; }
        float inv = 1.0f / sum;
        #pragma unroll
        for (int l = 0; l < NSH; ++l)
            s_attn[w][(hi * 16 + mrow) * NSH + l] = sc[l] * inv;
    }
    __syncthreads();

    // ---- pass 2: V projection per shift, weighted-accumulate O in registers ----
    v8f o0, o1;
    #pragma unroll
    for (int r = 0; r < 8; ++r) { o0[r] = 0.0f; o1[r] = 0.0f; }
    {
        const float* wb0 = &s_win[(64 + mrow) * 32];
        const float* wb1 = &s_win[(80 + mrow) * 32];
        float bv0 = s_bin[64 + mrow], bv1 = s_bin[80 + mrow];
        for (int l = 0; l < NSH; ++l) {
            int sel = s_sel[w][l * 16 + mrow];
            const float* vrow = kf + ((size_t)b * NPTS + (sel < 0 ? 0 : sel)) * CC;
            const float* prow = &s_pos[l * 32];
            v8f v0, v1;
            #pragma unroll
            for (int r = 0; r < 8; ++r) { v0[r] = bv0; v1[r] = bv1; }
            #pragma unroll
            for (int kk = 0; kk < 8; ++kk) {
                int kb = kk * 4 + hi * 2;
                v2f a;
                a[0] = (sel >= 0) ? (vrow[kb]     + prow[kb])     : 0.0f;
                a[1] = (sel >= 0) ? (vrow[kb + 1] + prow[kb + 1]) : 0.0f;
                v2f b0; b0[0] = wb0[kb]; b0[1] = wb0[kb + 1];
                v2f b1; b1[0] = wb1[kb]; b1[1] = wb1[kb + 1];
                v0 = wmma4(a, b0, v0);
                v1 = wmma4(a, b1, v1);
            }
            // column tile jt == head index; weight each row m with attn[m][jt][l]
            #pragma unroll
            for (int r = 0; r < 8; ++r) {
                int m = r + 8 * hi;
                o0[r] += s_attn[w][(0 * 16 + m) * NSH + l] * v0[r];
                o1[r] += s_attn[w][(1 * 16 + m) * NSH + l] * v1[r];
            }
        }
    }
    #pragma unroll
    for (int r = 0; r < 8; ++r) {
        int m = r + 8 * hi;
        s_o[w][m * 32 + mrow]      = o0[r];
        s_o[w][m * 32 + 16 + mrow] = o1[r];
    }
    __syncthreads();

    // ---- output projection: out = O @ out_w^T + out_b ----
    v8f f0, f1;
    {
        float bo0 = s_bout[mrow], bo1 = s_bout[16 + mrow];
        #pragma unroll
        for (int r = 0; r < 8; ++r) { f0[r] = bo0; f1[r] = bo1; }
        const float* ob0 = &s_wout[mrow * 32];
        const float* ob1 = &s_wout[(16 + mrow) * 32];
        const float* orow = &s_o[w][mrow * 32];
        #pragma unroll
        for (int kk = 0; kk < 8; ++kk) {
            int kb = kk * 4 + hi * 2;
            v2f a;  a[0] = orow[kb];  a[1] = orow[kb + 1];
            v2f b0; b0[0] = ob0[kb];  b0[1] = ob0[kb + 1];
            v2f b1; b1[0] = ob1[kb];  b1[1] = ob1[kb + 1];
            f0 = wmma4(a, b0, f0);
            f1 = wmma4(a, b1, f1);
        }
    }

    // ---- scatter to canvas (channel-major, unique pixels per batch) ----
    size_t obase = ((size_t)dir * BB + b) * CC * (size_t)HWSZ;
    #pragma unroll
    for (int r = 0; r < 8; ++r) {
        int m = r + 8 * hi;
        int pix = s_pix[w][m];
        if (pix >= 0) {
            out[obase + (size_t)mrow * HWSZ + pix]        = f0[r];
            out[obase + (size_t)(16 + mrow) * HWSZ + pix] = f1[r];
        }
    }
}

// ---------------- launch ----------------
extern "C" void kernel_launch(void* const* d_in, const int* in_sizes, int n_in,
                              void* d_out, int out_size, void* d_ws, size_t ws_size,
                              hipStream_t stream) {
    const float* li_f  = (const float*)d_in[0];
    const int*   li_c  = (const int*)d_in[1];
    const float* ra_f  = (const float*)d_in[2];
    const int*   ra_c  = (const int*)d_in[3];
    const float* pos   = (const float*)d_in[4];
    const float* inw1  = (const float*)d_in[5];
    const float* inb1  = (const float*)d_in[6];
    const float* outw1 = (const float*)d_in[7];
    const float* outb1 = (const float*)d_in[8];
    const float* inw2  = (const float*)d_in[9];
    const float* inb2  = (const float*)d_in[10];
    const float* outw2 = (const float*)d_in[11];
    const float* outb2 = (const float*)d_in[12];
    float* out = (float*)d_out;
    int* grid_ws = (int*)d_ws;
    (void)in_sizes; (void)n_in; (void)ws_size;

    // zero canvas (scattered pixels overwrite; rest must be 0 every call)
    hipMemsetAsync(d_out, 0, (size_t)out_size * sizeof(float), stream);
    // grid = -1 everywhere (0xFF bytes -> int -1)
    hipMemsetAsync(d_ws, 0xFF, (size_t)2 * BB * HWSZ * sizeof(int), stream);

    int tot = 2 * BB * NPTS;
    build_grid_kernel<<<(tot + 255) / 256, 256, 0, stream>>>(li_c, ra_c, grid_ws);

    dim3 g((NPTS + 63) / 64, BB, 2);
    fused_attn_kernel<<<g, 128, 0, stream>>>(li_f, li_c, ra_f, ra_c, pos,
                                             inw1, inb1, outw1, outb1,
                                             inw2, inb2, outw2, outb2,
                                             grid_ws, out);
}